// MSAN_83794811945592
// MI455X (gfx1250) — compile-verified
//
#include <hip/hip_runtime.h>
#include <hip/hip_bf16.h>

// CDNA5 / gfx1250 fused graph-aggregate + GEMM(+bias+ReLU) kernel.
// Uses V_WMMA_F32_16X16X4_F32 (exact f32 path) for the 12288x128x128 GEMM.

typedef float v2f __attribute__((ext_vector_type(2)));
typedef float v8f __attribute__((ext_vector_type(8)));

#define NN   12288
#define KK   32
#define DD   128
#define AS   132   // padded LDS row stride (132 % 64 == 4 -> conflict-free)

__global__ __launch_bounds__(256)
void MSAN_graphsage_fused(const int*   __restrict__ nodes,
                          const int*   __restrict__ neighbors,
                          const float* __restrict__ rawf,
                          const float* __restrict__ rw,
                          const float* __restrict__ W,
                          const float* __restrict__ bias,
                          float*       __restrict__ out)
{
    __shared__ __align__(16) float aggLds[16 * AS];   // 16 x 128 agg tile, padded
    __shared__ __align__(16) float wgtLds[16 * KK];   // gathered edge weights
    __shared__ __align__(16) int   nbrLds[16 * KK];   // neighbor ids

    const int t    = threadIdx.x;
    const int base = blockIdx.x * 16;                 // first node of this tile

    // ---- Phase 0: gather masked edge weights (the HBM-dominant part) ----
    // 512 (node,k) pairs over 256 threads -> 2 iterations.
    for (int i = t; i < 16 * KK; i += 256) {
        const int r   = i >> 5;          // node-in-tile
        const int k   = i & 31;
        const int b   = base + r;
        const int nid = nodes[b];
        const int nb  = neighbors[b * KK + k];
        nbrLds[i] = nb;
        wgtLds[i] = (nb == nid) ? 0.0f
                                : rw[(size_t)nid * (size_t)NN + (size_t)nb];
    }
    __syncthreads();

    // ---- Phase 1: agg[r][d] = sum_k w * raw_features[nb][d] ----
    // thread t: node r = t>>4, float8 chunk c = t&15 (d in [8c, 8c+8)).
    // 16 consecutive lanes of a chunk-row read a contiguous 512B feature row
    // via float4 loads (fully coalesced; raw_features is L2-resident, 6.3MB).
    {
        const int r     = t >> 4;
        const int dbase = (t & 15) * 8;
        float acc[8];
#pragma unroll
        for (int j = 0; j < 8; ++j) acc[j] = 0.0f;

#pragma unroll 4
        for (int k = 0; k < KK; ++k) {
            const float w  = wgtLds[r * KK + k];
            const int   nb = nbrLds[r * KK + k];
            const float4* fp = (const float4*)(rawf + (size_t)nb * DD + dbase);
            const float4 f0 = fp[0];
            const float4 f1 = fp[1];
            acc[0] += w * f0.x; acc[1] += w * f0.y;
            acc[2] += w * f0.z; acc[3] += w * f0.w;
            acc[4] += w * f1.x; acc[5] += w * f1.y;
            acc[6] += w * f1.z; acc[7] += w * f1.w;
        }
#pragma unroll
        for (int j = 0; j < 8; ++j) aggLds[r * AS + dbase + j] = acc[j];
    }
    __syncthreads();

    // ---- Phase 2: out_tile(16x128) = relu(agg(16x128) @ W^T + b) via WMMA ----
    // Wave wv owns output columns [16*wv, 16*wv+16). K=128 in 32 steps of 4.
    // ISA layouts (7.12.2, 32-bit):
    //   A (16x4): lane l holds A[l%16][(l>>4)*2 + v], v = VGPR 0/1
    //   B (4x16): lane l holds B[(l>>4)*2 + v][l%16];  B[k][n] = W_lin[n][k]
    //   D (16x16): VGPR j -> row j (lanes 0-15) / row j+8 (lanes 16-31)
    const int lane  = t & 31;
    const int wv    = t >> 5;
    const int nbase = wv * 16;
    const int mrow  = lane & 15;
    const int koff  = (lane >> 4) * 2;

    const float* Arow = aggLds + mrow * AS + koff;                      // LDS
    const float* Brow = W + (size_t)(nbase + mrow) * DD + koff;         // L2-hot

    v8f c = {};
#pragma unroll 4
    for (int kk = 0; kk < DD; kk += 4) {
        const float2 af = *(const float2*)(Arow + kk);   // ds_load_b64
        const float2 bf = *(const float2*)(Brow + kk);   // global_load_b64
        v2f a; a[0] = af.x; a[1] = af.y;
        v2f b; b[0] = bf.x; b[1] = bf.y;
        c = __builtin_amdgcn_wmma_f32_16x16x4_f32(
                /*neg_a=*/false, a, /*neg_b=*/false, b,
                /*c_mod=*/(short)0, c, /*reuse_a=*/false, /*reuse_b=*/false);
    }

    // ---- Store: bias + ReLU ----
    const int rr  = (lane < 16) ? 0 : 8;
    const int col = nbase + mrow;
    const float bb = bias[col];
#pragma unroll
    for (int j = 0; j < 8; ++j) {
        float v = c[j] + bb;
        v = v > 0.0f ? v : 0.0f;
        out[(size_t)(base + rr + j) * DD + col] = v;
    }
}

extern "C" void kernel_launch(void* const* d_in, const int* in_sizes, int n_in,
                              void* d_out, int out_size, void* d_ws, size_t ws_size,
                              hipStream_t stream)
{
    (void)in_sizes; (void)n_in; (void)out_size; (void)d_ws; (void)ws_size;
    const int*   nodes     = (const int*)  d_in[0];
    const int*   neighbors = (const int*)  d_in[1];
    const float* rawf      = (const float*)d_in[2];
    const float* rw        = (const float*)d_in[3];
    const float* W         = (const float*)d_in[4];
    const float* bias      = (const float*)d_in[5];
    float*       out       = (float*)d_out;

    dim3 grid(NN / 16);   // 768 tiles of 16 nodes
    dim3 block(256);      // 8 wave32s
    hipLaunchKernelGGL(MSAN_graphsage_fused, grid, block, 0, stream,
                       nodes, neighbors, rawf, rw, W, bias, out);
}